// MixtureOfLayersClassifier_13417477832998
// MI455X (gfx1250) — compile-verified
//
#include <hip/hip_runtime.h>
#include <hip/hip_bf16.h>
#include <math.h>

typedef float v2f __attribute__((ext_vector_type(2)));
typedef float v8f __attribute__((ext_vector_type(8)));

#define LN_EPS 1e-5f
#define MT 4   // M-subtiles (16 rows each) per wave -> 64x16 output per wave

// ---------------------------------------------------------------------------
// Pool: mean over HW for each (b,c) row. Block-per-row version for large HW.
// ---------------------------------------------------------------------------
__global__ void pool_mean_block(const float* __restrict__ x, float* __restrict__ out, int HW) {
    const int bc = blockIdx.x;
    const float* p = x + (size_t)bc * HW;
    float s = 0.f;
    if ((HW & 3) == 0) {
        const float4* p4 = (const float4*)p;
        const int n4 = HW >> 2;
        for (int i = threadIdx.x; i < n4; i += blockDim.x) {
            __builtin_prefetch(p4 + i + 4 * blockDim.x, 0, 1);
            float4 v = p4[i];
            s += (v.x + v.y) + (v.z + v.w);
        }
    } else {
        for (int i = threadIdx.x; i < HW; i += blockDim.x) s += p[i];
    }
    __shared__ float red[128];
    red[threadIdx.x] = s;
    __syncthreads();
    for (int off = 64; off > 0; off >>= 1) {
        if ((int)threadIdx.x < off) red[threadIdx.x] += red[threadIdx.x + off];
        __syncthreads();
    }
    if (threadIdx.x == 0) out[bc] = red[0] / (float)HW;
}

// Wave-per-row version for small HW (7x7, 14x14 maps).
__global__ void pool_mean_wave(const float* __restrict__ x, float* __restrict__ out,
                               int HW, int nrows) {
    const int wave = (int)(blockIdx.x * blockDim.x + threadIdx.x) >> 5;
    const int lane = threadIdx.x & 31;
    if (wave >= nrows) return;
    const float* p = x + (size_t)wave * HW;
    float s = 0.f;
    for (int i = lane; i < HW; i += 32) s += p[i];
    for (int off = 16; off > 0; off >>= 1) s += __shfl_down(s, off, 32);
    if (lane == 0) out[wave] = s / (float)HW;
}

// ---------------------------------------------------------------------------
// WMMA f32 GEMM: out[M,N] = act(A[M,K] @ W[K,N] + bias[N]); act 0=none 1=relu
// One wave computes a 64x16 tile (4 M-subtiles sharing each B fragment) with
// V_WMMA_F32_16X16X4_F32, stepping K by 4. Requires K%4==0 and M%64==0.
// N-edge handled with clamped addresses + multiplicative mask (no EXEC games).
// ---------------------------------------------------------------------------
__global__ void wmma_gemm_f32(const float* __restrict__ A, const float* __restrict__ W,
                              const float* __restrict__ bias, float* __restrict__ out,
                              int M, int N, int K, int act) {
    const int wave   = (int)(blockIdx.x * blockDim.x + threadIdx.x) >> 5;
    const int lane   = threadIdx.x & 31;
    const int tilesN = (N + 15) >> 4;
    const int grpsM  = M >> 6;                 // 64-row groups
    const int tm = wave / tilesN;
    const int tn = wave % tilesN;
    if (tm >= grpsM) return;                   // wave-uniform exit

    const int row0  = tm << 6;
    const int col0  = tn << 4;
    const int lhalf = lane & 15;               // 0..15
    const int hi    = lane >> 4;               // 0 or 1
    const int col   = col0 + lhalf;
    const bool cok  = (col < N);
    const int  colc = cok ? col : 0;           // clamped, always in-bounds
    const float msk = cok ? 1.f : 0.f;         // zero out-of-range columns

    v8f acc[MT] = {{}, {}, {}, {}};
    const float* a0 = A + (size_t)(row0 + lhalf) * K;

    for (int k = 0; k < K; k += 4) {
        const int ka = k + 2 * hi;             // even -> 8B-aligned within row
        // B fragment (4x16): rows ka, ka+1 at this lane's column (coalesced)
        v2f b;
        b.x = W[(size_t)ka * N + colc] * msk;
        b.y = W[(size_t)(ka + 1) * N + colc] * msk;
        #pragma unroll
        for (int t = 0; t < MT; ++t) {
            // A fragment (16x4): one aligned b64 load per subtile
            const float2 av = *(const float2*)(a0 + (size_t)(t << 4) * K + ka);
            v2f a; a.x = av.x; a.y = av.y;
            acc[t] = __builtin_amdgcn_wmma_f32_16x16x4_f32(
                false, a, false, b, (short)0, acc[t], false, false);
        }
    }

    if (cok) {
        const float bv = bias ? bias[col] : 0.f;
        #pragma unroll
        for (int t = 0; t < MT; ++t) {
            #pragma unroll
            for (int r = 0; r < 8; ++r) {
                const int row = row0 + (t << 4) + r + 8 * hi;  // D: VGPR r -> rows r, r+8
                float v = acc[t][r] + bv;
                if (act == 1) v = fmaxf(v, 0.f);
                out[(size_t)row * N + col] = v;
            }
        }
    }
}

// ---------------------------------------------------------------------------
// Gate head: gl[256,5] = h[256,512] @ Wg3[512,5] + bg3
// ---------------------------------------------------------------------------
__global__ void gate3_kernel(const float* __restrict__ h, const float* __restrict__ W,
                             const float* __restrict__ bias, float* __restrict__ gl) {
    const int idx = blockIdx.x * blockDim.x + threadIdx.x;
    if (idx >= 256 * 5) return;
    const int b = idx / 5, j = idx % 5;
    const float* hr = h + (size_t)b * 512;
    float s = bias[j];
    for (int k = 0; k < 512; ++k) s += hr[k] * W[k * 5 + j];
    gl[idx] = s;
}

// ---------------------------------------------------------------------------
// Per-row LayerNorm (ddof=0) + exact GELU, F=512, in-place. Block=256 threads.
// ---------------------------------------------------------------------------
__global__ void ln_gelu_kernel(float* __restrict__ h, const float* __restrict__ g,
                               const float* __restrict__ be) {
    const int b = blockIdx.x;
    float* row = h + (size_t)b * 512;
    const float x0 = row[threadIdx.x];
    const float x1 = row[threadIdx.x + 256];
    __shared__ float rs[256], rq[256];
    rs[threadIdx.x] = x0 + x1;
    rq[threadIdx.x] = x0 * x0 + x1 * x1;
    __syncthreads();
    for (int off = 128; off > 0; off >>= 1) {
        if ((int)threadIdx.x < off) {
            rs[threadIdx.x] += rs[threadIdx.x + off];
            rq[threadIdx.x] += rq[threadIdx.x + off];
        }
        __syncthreads();
    }
    const float mu  = rs[0] * (1.f / 512.f);
    const float var = rq[0] * (1.f / 512.f) - mu * mu;
    const float inv = rsqrtf(var + LN_EPS);
    #pragma unroll
    for (int t = 0; t < 2; ++t) {
        const int f = threadIdx.x + t * 256;
        const float y = (row[f] - mu) * inv * g[f] + be[f];
        row[f] = 0.5f * y * (1.f + erff(y * 0.70710678118654752f));
    }
}

// ---------------------------------------------------------------------------
// Top-2 softmax over gate logits + mixture of selected stage projections.
// stack layout: [stage][b][512]. Also emits gate_logits/topi/w outputs.
// ---------------------------------------------------------------------------
__global__ void topk_mix_kernel(const float* __restrict__ gl, const float* __restrict__ stack,
                                float* __restrict__ mixture,
                                float* __restrict__ out_gl, float* __restrict__ out_topi,
                                float* __restrict__ out_w) {
    const int b = blockIdx.x;
    __shared__ int si0, si1;
    __shared__ float sw0, sw1;
    if (threadIdx.x == 0) {
        float v[5];
        #pragma unroll
        for (int i = 0; i < 5; ++i) v[i] = gl[b * 5 + i];
        int i0 = 0;
        #pragma unroll
        for (int i = 1; i < 5; ++i) if (v[i] > v[i0]) i0 = i;       // first max (ties -> low idx)
        int i1 = (i0 == 0) ? 1 : 0;
        #pragma unroll
        for (int i = 0; i < 5; ++i) { if (i == i0) continue; if (v[i] > v[i1]) i1 = i; }
        const float m  = fmaxf(v[i0], v[i1]);
        const float e0 = __expf(v[i0] - m);
        const float e1 = __expf(v[i1] - m);
        const float inv = 1.f / (e0 + e1);
        si0 = i0; si1 = i1; sw0 = e0 * inv; sw1 = e1 * inv;
        #pragma unroll
        for (int i = 0; i < 5; ++i) out_gl[b * 5 + i] = v[i];
        out_topi[b * 2 + 0] = (float)i0;
        out_topi[b * 2 + 1] = (float)i1;
        out_w[b * 2 + 0] = e0 * inv;
        out_w[b * 2 + 1] = e1 * inv;
    }
    __syncthreads();
    const float w0 = sw0, w1 = sw1;
    const size_t r0 = ((size_t)si0 * 256 + b) * 512;
    const size_t r1 = ((size_t)si1 * 256 + b) * 512;
    for (int f = threadIdx.x; f < 512; f += blockDim.x)
        mixture[(size_t)b * 512 + f] = w0 * stack[r0 + f] + w1 * stack[r1 + f];
}

// ---------------------------------------------------------------------------
// Host-side orchestration
// ---------------------------------------------------------------------------
static inline int gemm_blocks(int M, int N) {
    const int tiles = (M >> 6) * ((N + 15) >> 4);   // 64-row groups x 16-col tiles
    return (tiles * 32 + 255) / 256;
}

extern "C" void kernel_launch(void* const* d_in, const int* in_sizes, int n_in,
                              void* d_out, int out_size, void* d_ws, size_t ws_size,
                              hipStream_t stream) {
    (void)in_sizes; (void)n_in; (void)out_size; (void)ws_size;
    const int B = 256, F = 512, H = 1024;

    const float* s[5];
    for (int i = 0; i < 5; ++i) s[i] = (const float*)d_in[i];
    const float* ff = (const float*)d_in[5];
    const float *Wp[5], *bp[5], *g[5], *be[5];
    for (int i = 0; i < 5; ++i) {
        Wp[i] = (const float*)d_in[6 + 4 * i + 0];
        bp[i] = (const float*)d_in[6 + 4 * i + 1];
        g[i]  = (const float*)d_in[6 + 4 * i + 2];
        be[i] = (const float*)d_in[6 + 4 * i + 3];
    }
    const float* Wg1 = (const float*)d_in[26]; const float* bg1 = (const float*)d_in[27];
    const float* Wg2 = (const float*)d_in[28]; const float* bg2 = (const float*)d_in[29];
    const float* Wg3 = (const float*)d_in[30]; const float* bg3 = (const float*)d_in[31];
    const float* Wc1 = (const float*)d_in[32]; const float* bc1 = (const float*)d_in[33];
    const float* Wc2 = (const float*)d_in[34]; const float* bc2 = (const float*)d_in[35];
    const float* Wc3 = (const float*)d_in[36]; const float* bc3 = (const float*)d_in[37];

    // Workspace layout (floats)
    float* ws = (float*)d_ws;
    const int chans[5] = {16, 24, 40, 80, 160};
    size_t off = 0;
    float* pooled_s[5];
    for (int i = 0; i < 5; ++i) { pooled_s[i] = ws + off; off += (size_t)B * chans[i]; }
    float* pooled_ff = ws + off; off += (size_t)B * 960;
    float* g_h1      = ws + off; off += (size_t)B * H;
    float* g_h2      = ws + off; off += (size_t)B * F;
    float* gate_log  = ws + off; off += (size_t)B * 5;
    float* stack     = ws + off; off += (size_t)5 * B * F;
    float* mixture   = ws + off; off += (size_t)B * F;
    float* c_h1      = ws + off; off += (size_t)B * H;
    float* c_h2      = ws + off; off += (size_t)B * F;

    // Output layout: logits[256*1000] | gate_logits[256*5] | topi[256*2] | w[256*2]
    float* out_logits = (float*)d_out;
    float* out_gl     = out_logits + (size_t)B * 1000;
    float* out_topi   = out_gl + (size_t)B * 5;
    float* out_w      = out_topi + (size_t)B * 2;

    // ---- 1. Pools (bandwidth-dominant stage, ~385 MB streamed) ----
    const int hw[5] = {112 * 112, 56 * 56, 28 * 28, 14 * 14, 7 * 7};
    for (int i = 0; i < 3; ++i)
        pool_mean_block<<<B * chans[i], 128, 0, stream>>>(s[i], pooled_s[i], hw[i]);
    for (int i = 3; i < 5; ++i) {
        const int rows = B * chans[i];
        pool_mean_wave<<<(rows * 32 + 255) / 256, 256, 0, stream>>>(s[i], pooled_s[i], hw[i], rows);
    }
    {
        const int rows = B * 960;
        pool_mean_wave<<<(rows * 32 + 255) / 256, 256, 0, stream>>>(ff, pooled_ff, 49, rows);
    }

    // ---- 2. Gate MLP (WMMA) ----
    wmma_gemm_f32<<<gemm_blocks(B, H), 256, 0, stream>>>(pooled_ff, Wg1, bg1, g_h1, B, H, 960, 1);
    wmma_gemm_f32<<<gemm_blocks(B, F), 256, 0, stream>>>(g_h1, Wg2, bg2, g_h2, B, F, H, 1);
    gate3_kernel<<<(B * 5 + 255) / 256, 256, 0, stream>>>(g_h2, Wg3, bg3, gate_log);

    // ---- 3. Per-stage projections (WMMA) + LayerNorm + GELU ----
    for (int i = 0; i < 5; ++i) {
        float* stage = stack + (size_t)i * B * F;
        wmma_gemm_f32<<<gemm_blocks(B, F), 256, 0, stream>>>(pooled_s[i], Wp[i], bp[i],
                                                             stage, B, F, chans[i], 0);
        ln_gelu_kernel<<<B, 256, 0, stream>>>(stage, g[i], be[i]);
    }

    // ---- 4. Top-2 softmax mixture (+ aux outputs) ----
    topk_mix_kernel<<<B, 256, 0, stream>>>(gate_log, stack, mixture, out_gl, out_topi, out_w);

    // ---- 5. Classifier MLP (WMMA); final GEMM writes logits to d_out ----
    wmma_gemm_f32<<<gemm_blocks(B, H), 256, 0, stream>>>(mixture, Wc1, bc1, c_h1, B, H, F, 1);
    wmma_gemm_f32<<<gemm_blocks(B, F), 256, 0, stream>>>(c_h1, Wc2, bc2, c_h2, B, F, H, 1);
    wmma_gemm_f32<<<gemm_blocks(B, 1000), 256, 0, stream>>>(c_h2, Wc3, bc3, out_logits,
                                                            B, 1000, F, 0);
}